// GConv_35639638622887
// MI455X (gfx1250) — compile-verified
//
#include <hip/hip_runtime.h>
#include <math.h>

// MI455X / gfx1250: wave32, WMMA f32 16x16x4 tensor-core FFT (four-step, 8192 = 64 x 128),
// with TDM (tensor_load_to_lds / tensor_store_from_lds) for the strided channel columns.

typedef __attribute__((ext_vector_type(2))) float v2f;
typedef __attribute__((ext_vector_type(8))) float v8f;
typedef unsigned int u32;
typedef unsigned long long u64;
typedef u32 u32x4 __attribute__((ext_vector_type(4)));
typedef u32 u32x8 __attribute__((ext_vector_type(8)));

#define SEQ_L 8192
#define NCH   96
#define NB    8
#define NTHR  256

static __device__ __forceinline__ v8f wmma4(v2f a, v2f b, v8f c) {
  // D(16x16,f32) = A(16x4,f32) * B(4x16,f32) + C   -> v_wmma_f32_16x16x4_f32
  return __builtin_amdgcn_wmma_f32_16x16x4_f32(false, a, false, b, (short)0, c, false, false);
}

static __device__ __forceinline__ void wait_tensor0() {
#if __has_builtin(__builtin_amdgcn_s_wait_tensorcnt)
  __builtin_amdgcn_s_wait_tensorcnt(0);
#else
  asm volatile("s_wait_tensorcnt 0" ::: "memory");
#endif
}

// ---- Tensor Data Mover: move one channel column (8192 f32, stride 96 f32) between
// global memory and contiguous LDS at byte offset lds_off. D# per CDNA5 ISA ch.8:
//   2D tile: tile_dim0=1, tile_dim1=8192, tensor_dim0_stride=96, data_size=4B.
//   Groups 2/3 present (4-group form) with benign dims = 1.
#define TDM_DESC(ga, lds_off)                                                          \
  u32x4 g0 = { 1u, (u32)(lds_off), (u32)(ga),                                          \
               (u32)(((ga) >> 32) & 0x1FFFFFFull) | (2u << 30) };                      \
  u32x8 g1 = { (2u << 16),            /* data_size=4B, wg_mask=0           */          \
               (96u << 16),           /* tensor_dim0 = 96 (lo16)           */          \
               (8192u << 16),         /* tensor_dim0 hi=0 | tensor_dim1 lo */          \
               (1u << 16),            /* tensor_dim1 hi=0 | tile_dim0 = 1  */          \
               8192u,                 /* tile_dim1 = 8192, tile_dim2 = 0   */          \
               96u,                   /* tensor_dim0_stride = 96 (lo32)    */          \
               0u, 0u };                                                               \
  u32x4 g2 = { 1u, 1u, 0u, 0u };      /* tensor_dim2=1, tensor_dim3=1      */          \
  u32x4 g3 = { 0u, (1u << 16), 0u, 0u } /* tensor_dim4=1 */

static __device__ __forceinline__ void tdm_load_col(const float* gptr, u32 lds_off) {
  const u64 ga = (u64)gptr;
  TDM_DESC(ga, lds_off);
  asm volatile("tensor_load_to_lds %0, %1, %2, %3"
               :: "s"(g0), "s"(g1), "s"(g2), "s"(g3) : "memory");
}

static __device__ __forceinline__ void tdm_store_col(float* gptr, u32 lds_off) {
  const u64 ga = (u64)gptr;
  TDM_DESC(ga, lds_off);
  asm volatile("tensor_store_from_lds %0, %1, %2, %3"
               :: "s"(g0), "s"(g1), "s"(g2), "s"(g3) : "memory");
}

// In-place 8192-point complex FFT (natural order in/out) on LDS arrays.
// Four-step: n = n1 + 64*n2, k = k2 + 128*k1.
//   Stage A: A[n1][k2] = sum_n2 x[n1+64*n2] * F128[k2][n2],  then *= W8192^(n1*k2)
//   Stage B: X[k2+128*k1] = sum_n1 F64[k1][n1] * A[n1][k2]
// 256 threads = 8 wave32s; wave w owns output column tile [16w,16w+16).
template<bool IMAG_ZERO>
static __device__ void fft8192(float* __restrict__ sxr, float* __restrict__ sxi, int tid) {
  const int lane = tid & 31;
  const int wv   = tid >> 5;
  const int nB   = wv << 4;
  const int lhi  = lane >> 4;
  const int llo  = lane & 15;
  const float TWO_PI = 6.28318530717958647692f;

  v8f Cr[4] = {};
  v8f Ci[4] = {};

  // ---------------- Stage A ----------------
  for (int kk = 0; kk < 32; ++kk) {
    const int kB = kk << 2;
    const int kA = kB + (lhi << 1);
    const int n_l = nB + llo;
    float s0, c0, s1, c1;
    __sincosf(TWO_PI * (float)((n_l * kA) & 127) * (1.0f / 128.0f), &s0, &c0);
    __sincosf(TWO_PI * (float)((n_l * (kA + 1)) & 127) * (1.0f / 128.0f), &s1, &c1);
    v2f Bc  = { c0,  c1  };
    v2f Bs  = { -s0, -s1 };   // Im(F128)
    v2f Bsn = { s0,  s1  };   // -Im(F128)  (f32 WMMA forbids A/B NEG bits)
#pragma unroll
    for (int mt = 0; mt < 4; ++mt) {
      const int m_l = (mt << 4) + llo;
      v2f Ar = { sxr[m_l + (kA << 6)], sxr[m_l + ((kA + 1) << 6)] };
      Cr[mt] = wmma4(Ar, Bc, Cr[mt]);
      Ci[mt] = wmma4(Ar, Bs, Ci[mt]);
      if (!IMAG_ZERO) {
        v2f Ai = { sxi[m_l + (kA << 6)], sxi[m_l + ((kA + 1) << 6)] };
        Cr[mt] = wmma4(Ai, Bsn, Cr[mt]);
        Ci[mt] = wmma4(Ai, Bc,  Ci[mt]);
      }
    }
  }
  __syncthreads();   // all reads of x done -> safe to overwrite in place
#pragma unroll
  for (int mt = 0; mt < 4; ++mt) {
#pragma unroll
    for (int r = 0; r < 8; ++r) {
      const int m = (mt << 4) + r + (lhi << 3);   // n1
      const int n = nB + llo;                     // k2
      float s, c;
      __sincosf(TWO_PI * (float)(m * n) * (1.0f / 8192.0f), &s, &c);
      const float re = Cr[mt][r] * c + Ci[mt][r] * s;
      const float im = Ci[mt][r] * c - Cr[mt][r] * s;
      sxr[(m << 7) + n] = re;
      sxi[(m << 7) + n] = im;
    }
  }
  __syncthreads();

  // ---------------- Stage B ----------------
#pragma unroll
  for (int mt = 0; mt < 4; ++mt) { Cr[mt] = (v8f){}; Ci[mt] = (v8f){}; }
  for (int kk = 0; kk < 16; ++kk) {
    const int kB = kk << 2;
    const int kA = kB + (lhi << 1);
    const int n_l = nB + llo;
    v2f Br = { sxr[(kA << 7) + n_l], sxr[((kA + 1) << 7) + n_l] };
    v2f Bi = { sxi[(kA << 7) + n_l], sxi[((kA + 1) << 7) + n_l] };
#pragma unroll
    for (int mt = 0; mt < 4; ++mt) {
      const int m_l = (mt << 4) + llo;            // k1
      float s0, c0, s1, c1;
      __sincosf(TWO_PI * (float)((m_l * kA) & 63) * (1.0f / 64.0f), &s0, &c0);
      __sincosf(TWO_PI * (float)((m_l * (kA + 1)) & 63) * (1.0f / 64.0f), &s1, &c1);
      v2f Ac  = { c0,  c1  };
      v2f As  = { -s0, -s1 };
      v2f Asn = { s0,  s1  };
      Cr[mt] = wmma4(Ac,  Br, Cr[mt]);
      Cr[mt] = wmma4(Asn, Bi, Cr[mt]);
      Ci[mt] = wmma4(Ac,  Bi, Ci[mt]);
      Ci[mt] = wmma4(As,  Br, Ci[mt]);
    }
  }
  __syncthreads();   // all reads of a[] done
#pragma unroll
  for (int mt = 0; mt < 4; ++mt) {
#pragma unroll
    for (int r = 0; r < 8; ++r) {
      const int k1 = (mt << 4) + r + (lhi << 3);
      const int k2 = nB + llo;
      const int idx = k2 + (k1 << 7);
      sxr[idx] = Cr[mt][r];
      sxi[idx] = Ci[mt][r];
    }
  }
  __syncthreads();
}

// Multi-scale bilinear kernel build: k[t, d].
static __device__ __forceinline__ float build_k(const float* __restrict__ kern, int d, int t) {
  int i = 0;
  while (t >= 64 * ((1 << (i + 1)) - 1)) ++i;
  const int   j     = t - 64 * ((1 << i) - 1);
  const float inv   = __builtin_ldexpf(1.0f, -i);
  const float coord = ((float)j + 0.5f) * inv - 0.5f;
  const float f0    = floorf(coord);
  const float fr    = coord - f0;
  int i0 = (int)f0;
  int i1 = i0 + 1;
  i0 = i0 < 0 ? 0 : (i0 > 63 ? 63 : i0);
  i1 = i1 < 0 ? 0 : (i1 > 63 ? 63 : i1);
  const float v = (1.0f - fr) * kern[i0 * NCH + d] + fr * kern[i1 * NCH + d];
  return v * inv;
}

// Pass 1: build k[:, d], FFT, write K_f to workspace ([d][f], contiguous, coalesced stores).
__global__ __launch_bounds__(NTHR) void kf_kernel(const float* __restrict__ kern,
                                                  float* __restrict__ kfr,
                                                  float* __restrict__ kfi) {
  __shared__ float smem[2 * SEQ_L];
  float* sxr = smem;
  float* sxi = smem + SEQ_L;
  const int d = blockIdx.x;
  const int tid = threadIdx.x;
  for (int t = tid; t < SEQ_L; t += NTHR) { sxr[t] = build_k(kern, d, t); sxi[t] = 0.0f; }
  __syncthreads();
  fft8192<true>(sxr, sxi, tid);
  for (int f = tid; f < SEQ_L; f += NTHR) {
    kfr[(size_t)d * SEQ_L + f] = sxr[f];
    kfi[(size_t)d * SEQ_L + f] = sxi[f];
  }
}

// Pass 2: per (b, d): TDM-load u -> FFT -> * K_f -> conj -> FFT -> Re/8192 + bias -> TDM-store.
__global__ __launch_bounds__(NTHR) void conv_kernel(const float* __restrict__ sig,
                                                    const float* __restrict__ bias,
                                                    const float* __restrict__ kfr,
                                                    const float* __restrict__ kfi,
                                                    float* __restrict__ out) {
  __shared__ float smem[2 * SEQ_L];
  float* sxr = smem;                 // LDS byte offset 0 (TDM target)
  float* sxi = smem + SEQ_L;
  const int b = blockIdx.x / NCH;
  const int d = blockIdx.x % NCH;
  const int tid = threadIdx.x;

  const float* u = sig + (size_t)b * SEQ_L * NCH + d;
  if (tid < 32) tdm_load_col(u, 0u);                    // DMA the strided column into sxr
  for (int t = tid; t < SEQ_L; t += NTHR) sxi[t] = 0.0f; // overlap with DMA
  if (tid < 32) wait_tensor0();
  __syncthreads();

  fft8192<true>(sxr, sxi, tid);

  const float* krp = kfr + (size_t)d * SEQ_L;
  const float* kip = kfi + (size_t)d * SEQ_L;
  for (int f = tid; f < SEQ_L; f += NTHR) {
    const float ur = sxr[f], ui = sxi[f];
    const float kr = krp[f], ki = kip[f];
    sxr[f] = ur * kr - ui * ki;          // Re(U*K)
    sxi[f] = -(ur * ki + ui * kr);       // conj: inverse via forward FFT
  }
  __syncthreads();

  fft8192<false>(sxr, sxi, tid);

  const float bv = bias[d];
  for (int t = tid; t < SEQ_L; t += NTHR) {
    sxr[t] = sxr[t] * (1.0f / 8192.0f) + bv;   // finalize in LDS
  }
  __syncthreads();
  if (tid < 32) {
    tdm_store_col(out + (size_t)b * SEQ_L * NCH + d, 0u);  // DMA scatter back
    wait_tensor0();
  }
}

// Fallback if workspace too small: K_f per WG in 64 VGPRs/thread; u-DMA overlaps zero-fill.
__global__ __launch_bounds__(NTHR) void conv_fused_kernel(const float* __restrict__ sig,
                                                          const float* __restrict__ kern,
                                                          const float* __restrict__ bias,
                                                          float* __restrict__ out) {
  __shared__ float smem[2 * SEQ_L];
  float* sxr = smem;
  float* sxi = smem + SEQ_L;
  const int b = blockIdx.x / NCH;
  const int d = blockIdx.x % NCH;
  const int tid = threadIdx.x;

  for (int t = tid; t < SEQ_L; t += NTHR) { sxr[t] = build_k(kern, d, t); sxi[t] = 0.0f; }
  __syncthreads();
  fft8192<true>(sxr, sxi, tid);

  float kfrr[32], kfir[32];
#pragma unroll
  for (int q = 0; q < 32; ++q) { const int f = tid + NTHR * q; kfrr[q] = sxr[f]; kfir[q] = sxi[f]; }
  __syncthreads();                                       // K_f safely in registers

  const float* u = sig + (size_t)b * SEQ_L * NCH + d;
  if (tid < 32) tdm_load_col(u, 0u);
  for (int t = tid; t < SEQ_L; t += NTHR) sxi[t] = 0.0f;
  if (tid < 32) wait_tensor0();
  __syncthreads();

  fft8192<true>(sxr, sxi, tid);
#pragma unroll
  for (int q = 0; q < 32; ++q) {
    const int f = tid + NTHR * q;
    const float ur = sxr[f], ui = sxi[f];
    sxr[f] = ur * kfrr[q] - ui * kfir[q];
    sxi[f] = -(ur * kfir[q] + ui * kfrr[q]);
  }
  __syncthreads();
  fft8192<false>(sxr, sxi, tid);

  const float bv = bias[d];
  for (int t = tid; t < SEQ_L; t += NTHR) {
    sxr[t] = sxr[t] * (1.0f / 8192.0f) + bv;
  }
  __syncthreads();
  if (tid < 32) {
    tdm_store_col(out + (size_t)b * SEQ_L * NCH + d, 0u);
    wait_tensor0();
  }
}

extern "C" void kernel_launch(void* const* d_in, const int* in_sizes, int n_in,
                              void* d_out, int out_size, void* d_ws, size_t ws_size,
                              hipStream_t stream) {
  (void)in_sizes; (void)n_in; (void)out_size;
  const float* sig  = (const float*)d_in[0];   // (8, 8192, 96) f32
  const float* kern = (const float*)d_in[1];   // (64, 96) f32
  const float* bias = (const float*)d_in[2];   // (96,) f32
  float* out = (float*)d_out;                  // (8, 8192, 96) f32

  const size_t need = (size_t)2 * NCH * SEQ_L * sizeof(float);  // ~6.3 MB
  if (d_ws != nullptr && ws_size >= need) {
    float* kfr = (float*)d_ws;
    float* kfi = kfr + (size_t)NCH * SEQ_L;
    kf_kernel<<<NCH, NTHR, 0, stream>>>(kern, kfr, kfi);
    conv_kernel<<<NB * NCH, NTHR, 0, stream>>>(sig, bias, kfr, kfi, out);
  } else {
    conv_fused_kernel<<<NB * NCH, NTHR, 0, stream>>>(sig, kern, bias, out);
  }
}